// TPLayer_81466939671104
// MI455X (gfx1250) — compile-verified
//
#include <hip/hip_runtime.h>
#include <stdint.h>

#define MAX_OFFSET 8

// clang ext vectors (support .x/.y/.z/.w and lower to b128 ds/global ops)
typedef __attribute__((ext_vector_type(4))) float    f32x4;
typedef __attribute__((ext_vector_type(4))) int      s32x4;
typedef __attribute__((ext_vector_type(8))) int      s32x8;
typedef __attribute__((ext_vector_type(4))) unsigned u32x4;

#if __has_include(<hip/amd_detail/amd_gfx1250_TDM.h>)
#define TDM_ARGS6 1   // therock/clang-23 toolchain: 6-arg tensor builtin
#endif

namespace {
constexpr int kB = 32, kC = 64, kH = 160, kW = 160;
constexpr int kWQ      = kW / 4;              // 40 float4 per row
constexpr int kPlane   = kH * kW;             // 25600 elements per (b,c) plane
constexpr int kPlaneQ  = kH * kWQ;            // 6400 float4 per plane
constexpr int kThreads = 256;                 // 8 wave32s
constexpr int kNCH     = 4;                   // channels per block (masks amortized 4x)
constexpr int kTI      = 40;                  // output rows per block (H/kTI = 4 strips)
constexpr int kSRows   = kTI + MAX_OFFSET - 1;// 47 staged rows per channel (7-row halo)
constexpr int kSlotQ   = kSRows * kWQ;        // 1880 float4 per channel slot
constexpr int kTileQ   = kTI * kWQ;           // 1600 output quads per channel
}

__device__ __forceinline__ int imin_(int a, int b) { return a < b ? a : b; }

__global__ __launch_bounds__(kThreads)
void top_pool_tdm(const float* __restrict__ x,
                  const int*   __restrict__ offset,
                  float*       __restrict__ out)
{
  // 4 channel slots of 47 rows: 4*1880*16 = 120,320 B (2 blocks/WGP of 320 KiB)
  // Deepest read: 3*1880 + 1599 + 7*40 = 7519 < 7520 -> no address clamping needed.
  __shared__ f32x4 smem[kNCH * kSlotQ];

  const int bid   = blockIdx.x;        // 2048 blocks = B * (C/4) * (H/40)
  const int strip = bid & 3;           // row strip
  const int cg    = (bid >> 2) & 15;   // channel group
  const int b     = bid >> 6;          // batch
  const int i0    = strip * kTI;
  const int c0    = cg * kNCH;
  const int rs    = (kH - i0) < kSRows ? (kH - i0) : kSRows; // rows to DMA (47, last strip 40)

  // ---------- TDM: one 2-D descriptor per channel (wave 0 issues all four) ----------
  const unsigned lds0 = (unsigned)(uintptr_t)(void*)&smem[0];

  // D# group 1 (shared): data_size=4B, tensor WxH row-major, tile = W x rs
  s32x8 g1 = { (int)(2u << 16),                 // [17:16] data_size = 4 bytes
               (int)((unsigned)kW << 16),       // tensor_dim0[15:0] = W
               (int)((unsigned)kH << 16),       // tensor_dim1[15:0] = H
               (int)((unsigned)kW << 16),       // tile_dim0 = W
               rs,                              // tile_dim1 = rs | tile_dim2 = 0
               kW, 0, 0 };                      // tensor_dim0_stride = W
  s32x4 gz4 = { 0, 0, 0, 0 };

  if (threadIdx.x < 32) {                // whole-wave branch: only wave 0 issues DMAs
#pragma unroll
    for (int ch = 0; ch < kNCH; ++ch) {
      const unsigned long long ga = (unsigned long long)(uintptr_t)(const void*)
          (x + (size_t)(b * kC + c0 + ch) * kPlane + (size_t)i0 * kW);
      u32x4 g0 = { 1u,                                        // count=1
                   lds0 + (unsigned)(ch * kSlotQ * 16),       // LDS slot base
                   (unsigned)ga,                              // global addr [31:0]
                   ((unsigned)(ga >> 32) & 0x01FFFFFFu) | (2u << 30) }; // [56:32] | type=2
#ifdef TDM_ARGS6
      s32x8 gz8 = { 0, 0, 0, 0, 0, 0, 0, 0 };
      __builtin_amdgcn_tensor_load_to_lds(g0, g1, gz4, gz4, gz8, 0);
#else
      __builtin_amdgcn_tensor_load_to_lds(g0, g1, gz4, gz4, 0);
#endif
    }
    __builtin_amdgcn_s_wait_tensorcnt(0);  // TENSORcnt is per-wave
  }
  __syncthreads();                         // publish LDS tiles to all 8 waves

  // ---------- windowed max: masks computed once, applied to 4 channels ----------
  const s32x4* __restrict__ offq =
      (const s32x4*)offset + (size_t)b * kPlaneQ + (size_t)i0 * kWQ;
  f32x4* __restrict__ outq =
      (f32x4*)out + (size_t)(b * kC + c0) * kPlaneQ + (size_t)i0 * kWQ;

  for (int k = (int)threadIdx.x; k < kTileQ; k += kThreads) {
    const int r  = k / kWQ;                // local row in strip
    const int om = (kH - 1) - (i0 + r);    // max legal downward reach
    const s32x4 o4 = offq[k];              // per-pixel offsets (channel-independent)
    const int o0 = imin_(o4.x, om);
    const int o1 = imin_(o4.y, om);
    const int o2 = imin_(o4.z, om);
    const int o3 = imin_(o4.w, om);

    f32x4 m[kNCH];
#pragma unroll
    for (int ch = 0; ch < kNCH; ++ch) m[ch] = smem[ch * kSlotQ + k];  // d = 0

#pragma unroll
    for (int d = 1; d < MAX_OFFSET; ++d) {
      const bool c0m = (d <= o0), c1m = (d <= o1), c2m = (d <= o2), c3m = (d <= o3);
#pragma unroll
      for (int ch = 0; ch < kNCH; ++ch) {
        const f32x4 v = smem[ch * kSlotQ + k + d * kWQ];   // ds_read_b128
        m[ch].x = c0m ? fmaxf(m[ch].x, v.x) : m[ch].x;     // masks shared across channels
        m[ch].y = c1m ? fmaxf(m[ch].y, v.y) : m[ch].y;
        m[ch].z = c2m ? fmaxf(m[ch].z, v.z) : m[ch].z;
        m[ch].w = c3m ? fmaxf(m[ch].w, v.w) : m[ch].w;
      }
    }
#pragma unroll
    for (int ch = 0; ch < kNCH; ++ch) {
      __builtin_nontemporal_store(m[ch], &outq[(size_t)ch * kPlaneQ + k]); // streaming b128
    }
  }
}

extern "C" void kernel_launch(void* const* d_in, const int* in_sizes, int n_in,
                              void* d_out, int out_size, void* d_ws, size_t ws_size,
                              hipStream_t stream) {
  const float* x   = (const float*)d_in[0];
  const int*   off = (const int*)d_in[1];
  float*       out = (float*)d_out;
  (void)in_sizes; (void)n_in; (void)out_size; (void)d_ws; (void)ws_size;

  // B * (C/kNCH) * (H/kTI) = 32 * 16 * 4 = 2048 blocks
  top_pool_tdm<<<dim3(kB * (kC / kNCH) * (kH / kTI)), dim3(kThreads), 0, stream>>>(x, off, out);
}